// GNNClassifier_28587302322870
// MI455X (gfx1250) — compile-verified
//
#include <hip/hip_runtime.h>
#include <hip/hip_bf16.h>
#include <stdint.h>

typedef float v2f __attribute__((ext_vector_type(2)));
typedef float v8f __attribute__((ext_vector_type(8)));

#define N_NODES 50000
#define N_EDGES 400000
#define DIM_IN  128
#define DIM_H   512
#define NUM_CLS 128
#define ENC_NEG_INF 0x007FFFFFu

// ---- order-preserving float <-> u32 encoding (exact float max via u32 atomicMax) ----
__device__ __forceinline__ unsigned flipf(float f) {
    unsigned u = __float_as_uint(f);
    return u ^ ((unsigned)((int)u >> 31) | 0x80000000u);
}
__device__ __forceinline__ float unflipf(unsigned u) {
    unsigned b = (u & 0x80000000u) ? (u ^ 0x80000000u) : ~u;
    return __uint_as_float(b);
}
__device__ __forceinline__ float decode_agg(unsigned u) {
    // flip(-inf) sentinel == empty segment -> 0.0 (matches isfinite fixup)
    return (u == ENC_NEG_INF) ? 0.0f : unflipf(u);
}

// ---- edge weights: ew = 1/(pos[src]-pos[dst]) ----
__global__ void ew_kernel(const int* __restrict__ src, const int* __restrict__ dst,
                          const float* __restrict__ pos, float* __restrict__ ew, int E) {
    int i = blockIdx.x * blockDim.x + threadIdx.x;
    if (i < E) ew[i] = 1.0f / (pos[src[i]] - pos[dst[i]]);
}

// ---- fill u32 buffer with sentinel (vectorized) ----
__global__ void fill_u32_kernel(unsigned* __restrict__ p, unsigned v, int n4) {
    int i = blockIdx.x * blockDim.x + threadIdx.x;
    if (i < n4) {
        uint4 q; q.x = v; q.y = v; q.z = v; q.w = v;
        ((uint4*)p)[i] = q;
    }
}

// ---- scatter-max: agg[dst] = max(agg[dst], x[src]*ew) over edges, 4 feats/thread ----
__global__ void scatter_max_kernel(const float* __restrict__ X,
                                   const int* __restrict__ src, const int* __restrict__ dst,
                                   const float* __restrict__ ew, unsigned* __restrict__ agg,
                                   int E, int D) {
    const int D4  = D >> 2;
    const int epb = blockDim.x / D4;                 // edges per block
    const int e   = blockIdx.x * epb + threadIdx.x / D4;
    const int c   = (threadIdx.x % D4) << 2;
    if (e >= E) return;
    const int s = src[e], d = dst[e];
    const float w = ew[e];
    const float4 v = *(const float4*)(X + (size_t)s * D + c);
    unsigned* q = agg + (size_t)d * D + c;
    atomicMax(q + 0, flipf(v.x * w));
    atomicMax(q + 1, flipf(v.y * w));
    atomicMax(q + 2, flipf(v.z * w));
    atomicMax(q + 3, flipf(v.w * w));
}

// ---- fused dual GEMM: C = dec(A1)@W1^T + A2@W2^T + bias, W row-major [Ncols, K] ----
// block = 256 thr = 8 waves; tile = 32 rows x 128 cols; wave w -> two 16x16 WMMA tiles
// (dual accumulators sharing one B fragment: 2x WMMA dependency distance, half the
//  B-LDS traffic per FLOP). K1, K2 are multiples of 64 (or K1 == 0) -> uniform source
// per 64-wide K-chunk.
#define KC 64
#define LDST 66
__global__ void gemm_wmma_kernel(const unsigned* __restrict__ A1, const float* __restrict__ A2,
                                 const float* __restrict__ W1, const float* __restrict__ W2,
                                 const float* __restrict__ bias, float* __restrict__ C,
                                 int M, int Ncols, int K1, int K2) {
    __shared__ float sA[32 * LDST];
    __shared__ float sB[128 * LDST];

    const int tid  = threadIdx.x;
    const int wave = tid >> 5;
    const int lane = tid & 31;
    const int h    = lane >> 4;     // K-half selector (fp32 WMMA A/B layout)
    const int lr   = lane & 15;     // row (A) / col (B) within fragment
    const int rowBase = blockIdx.x * 32;
    const int colBase = blockIdx.y * 128;
    const int KT = K1 + K2;

    v8f acc0 = {};
    v8f acc1 = {};

    // A cooperative-load mapping: 32 rows x 64 k = 2048 elems / 256 thr = 8 each
    const int ar = tid >> 3;            // 0..31
    const int ak = (tid & 7) << 3;      // 0,8,...,56
    // B cooperative-load mapping: 128 cols x 64 k = 8192 elems / 256 thr = 32 each
    const int bc = tid >> 1;            // 0..127
    const int bk = (tid & 1) << 5;      // 0 or 32
    const int rowA = (rowBase + ar < M) ? (rowBase + ar) : (M - 1);  // tail clamp
    const int colB = colBase + bc;

    for (int kc = 0; kc < KT; kc += KC) {
        const bool first = (kc < K1);
        // ---- stage A tile (decode u32-encoded max-agg, or raw float) ----
        {
            float f[8];
            if (first) {
                const uint4 e0 = *(const uint4*)(A1 + (size_t)rowA * K1 + kc + ak);
                const uint4 e1 = *(const uint4*)(A1 + (size_t)rowA * K1 + kc + ak + 4);
                f[0] = decode_agg(e0.x); f[1] = decode_agg(e0.y);
                f[2] = decode_agg(e0.z); f[3] = decode_agg(e0.w);
                f[4] = decode_agg(e1.x); f[5] = decode_agg(e1.y);
                f[6] = decode_agg(e1.z); f[7] = decode_agg(e1.w);
            } else {
                const float4 g0 = *(const float4*)(A2 + (size_t)rowA * K2 + (kc - K1) + ak);
                const float4 g1 = *(const float4*)(A2 + (size_t)rowA * K2 + (kc - K1) + ak + 4);
                f[0] = g0.x; f[1] = g0.y; f[2] = g0.z; f[3] = g0.w;
                f[4] = g1.x; f[5] = g1.y; f[6] = g1.z; f[7] = g1.w;
            }
            float* a = &sA[ar * LDST + ak];
#pragma unroll
            for (int j = 0; j < 8; ++j) a[j] = f[j];
        }
        // ---- stage B tile (B = W^T: sB[col][k] = W[col][k]) ----
        {
            const float* Wp = first ? (W1 + (size_t)colB * K1 + kc + bk)
                                    : (W2 + (size_t)colB * K2 + (kc - K1) + bk);
            float* bdst = &sB[bc * LDST + bk];
#pragma unroll
            for (int j = 0; j < 8; ++j) {
                const float4 fq = *(const float4*)(Wp + j * 4);
                bdst[j * 4 + 0] = fq.x; bdst[j * 4 + 1] = fq.y;
                bdst[j * 4 + 2] = fq.z; bdst[j * 4 + 3] = fq.w;
            }
        }
        // prefetch next weight chunk into GL2 while we compute this one
        if (kc + KC < KT) {
            const bool nf = (kc + KC < K1);
            const float* Wn = nf ? (W1 + (size_t)colB * K1 + (kc + KC) + bk)
                                 : (W2 + (size_t)colB * K2 + (kc + KC - K1) + bk);
            __builtin_prefetch(Wn, 0, 1);
        }
        __syncthreads();

        // ---- 32 x V_WMMA_F32_16X16X4_F32 over the chunk (two interleaved acc chains) ----
        // A frag lane mapping (ISA 7.12.2): lanes0-15 K={kk,kk+1}, lanes16-31 K={kk+2,kk+3}
        const float* pa0 = &sA[lr * LDST + 2 * h];
        const float* pa1 = &sA[(16 + lr) * LDST + 2 * h];
        const float* pb  = &sB[(wave * 16 + lr) * LDST + 2 * h];
#pragma unroll
        for (int kk = 0; kk < KC; kk += 4) {
            const v2f a0 = *(const v2f*)(pa0 + kk);
            const v2f a1 = *(const v2f*)(pa1 + kk);
            const v2f b  = *(const v2f*)(pb + kk);
            acc0 = __builtin_amdgcn_wmma_f32_16x16x4_f32(
                false, a0, false, b, (short)0, acc0, false, false);
            acc1 = __builtin_amdgcn_wmma_f32_16x16x4_f32(
                false, a1, false, b, (short)0, acc1, false, false);
        }
        __syncthreads();
    }

    // ---- epilogue: +bias, store (C layout: acc[v] -> row = v + 8*h, col = lr) ----
    const int col = colBase + wave * 16 + lr;
    const float bv = bias[col];
#pragma unroll
    for (int v = 0; v < 8; ++v) {
        const int r0 = rowBase + v + 8 * h;
        const int r1 = rowBase + 16 + v + 8 * h;
        if (r0 < M) C[(size_t)r0 * Ncols + col] = acc0[v] + bv;
        if (r1 < M) C[(size_t)r1 * Ncols + col] = acc1[v] + bv;
    }
}

// ---- LayerNorm(512) + ReLU in place; one wave32 per row ----
__global__ void ln_relu_kernel(float* __restrict__ H, const float* __restrict__ g,
                               const float* __restrict__ b, int M) {
    const int wave = threadIdx.x >> 5;
    const int lane = threadIdx.x & 31;
    const int row  = blockIdx.x * 8 + wave;
    if (row >= M) return;
    float* p = H + (size_t)row * DIM_H;
    float4 v[4];
    float s = 0.f, ss = 0.f;
#pragma unroll
    for (int i = 0; i < 4; ++i) {
        v[i] = *(const float4*)(p + lane * 4 + i * 128);
        s  += v[i].x + v[i].y + v[i].z + v[i].w;
        ss += v[i].x * v[i].x + v[i].y * v[i].y + v[i].z * v[i].z + v[i].w * v[i].w;
    }
#pragma unroll
    for (int off = 16; off >= 1; off >>= 1) {
        s  += __shfl_xor(s, off, 32);
        ss += __shfl_xor(ss, off, 32);
    }
    const float mean = s * (1.0f / DIM_H);
    float var = ss * (1.0f / DIM_H) - mean * mean;
    var = var < 0.f ? 0.f : var;
    const float rstd = rsqrtf(var + 1e-5f);
#pragma unroll
    for (int i = 0; i < 4; ++i) {
        const int c = lane * 4 + i * 128;
        const float4 gv = *(const float4*)(g + c);
        const float4 bv = *(const float4*)(b + c);
        float4 o;
        o.x = fmaxf(0.f, (v[i].x - mean) * rstd * gv.x + bv.x);
        o.y = fmaxf(0.f, (v[i].y - mean) * rstd * gv.y + bv.y);
        o.z = fmaxf(0.f, (v[i].z - mean) * rstd * gv.z + bv.z);
        o.w = fmaxf(0.f, (v[i].w - mean) * rstd * gv.w + bv.w);
        *(float4*)(p + c) = o;
    }
}

extern "C" void kernel_launch(void* const* d_in, const int* in_sizes, int n_in,
                              void* d_out, int out_size, void* d_ws, size_t ws_size,
                              hipStream_t stream) {
    const float* x      = (const float*)d_in[0];
    const int*   ei     = (const int*)  d_in[1];
    const float* pos    = (const float*)d_in[2];
    const float* Wrel1  = (const float*)d_in[3];
    const float* brel1  = (const float*)d_in[4];
    const float* Wroot1 = (const float*)d_in[5];
    const float* g1     = (const float*)d_in[6];
    const float* b1     = (const float*)d_in[7];
    const float* Wrel2  = (const float*)d_in[8];
    const float* brel2  = (const float*)d_in[9];
    const float* Wroot2 = (const float*)d_in[10];
    const float* g2     = (const float*)d_in[11];
    const float* b2     = (const float*)d_in[12];
    const float* Wcls   = (const float*)d_in[13];
    const float* bcls   = (const float*)d_in[14];
    const int* src = ei;
    const int* dst = ei + N_EDGES;
    float* out = (float*)d_out;

    // workspace layout (bytes): ew | h1 | agg2 | {agg1 (early) / h2 (late)}
    char* ws = (char*)d_ws;
    float*    ew   = (float*)   (ws);                 //   1.6 MB
    float*    h1   = (float*)   (ws + 1600000);       // 102.4 MB
    unsigned* agg2 = (unsigned*)(ws + 104000000);     // 102.4 MB
    unsigned* agg1 = (unsigned*)(ws + 206400000);     //  25.6 MB (dead before h2)
    float*    h2   = (float*)   (ws + 206400000);     // 102.4 MB

    const int gridM = (N_NODES + 31) / 32;            // 1563 (last tile half-full)

    // 1. edge weights
    ew_kernel<<<(N_EDGES + 255) / 256, 256, 0, stream>>>(src, dst, pos, ew, N_EDGES);

    // 2. layer 1: scatter-max of x*ew into agg1 (encoded)
    fill_u32_kernel<<<(N_NODES * DIM_IN / 4 + 255) / 256, 256, 0, stream>>>(
        agg1, ENC_NEG_INF, N_NODES * DIM_IN / 4);
    scatter_max_kernel<<<N_EDGES / 8, 256, 0, stream>>>(x, src, dst, ew, agg1,
                                                        N_EDGES, DIM_IN);
    // 3. h1 = dec(agg1)@Wrel1^T + x@Wroot1^T + brel1  (K = 128+128)
    gemm_wmma_kernel<<<dim3(gridM, DIM_H / 128), 256, 0, stream>>>(
        agg1, x, Wrel1, Wroot1, brel1, h1, N_NODES, DIM_H, DIM_IN, DIM_IN);
    ln_relu_kernel<<<N_NODES / 8, 256, 0, stream>>>(h1, g1, b1, N_NODES);

    // 4. layer 2: scatter-max of h1*ew into agg2 (encoded)
    fill_u32_kernel<<<(N_NODES * DIM_H / 4 + 255) / 256, 256, 0, stream>>>(
        agg2, ENC_NEG_INF, N_NODES * DIM_H / 4);
    scatter_max_kernel<<<N_EDGES / 2, 256, 0, stream>>>(h1, src, dst, ew, agg2,
                                                        N_EDGES, DIM_H);
    // 5. h2 = dec(agg2)@Wrel2^T + h1@Wroot2^T + brel2  (K = 512+512)
    gemm_wmma_kernel<<<dim3(gridM, DIM_H / 128), 256, 0, stream>>>(
        agg2, h1, Wrel2, Wroot2, brel2, h2, N_NODES, DIM_H, DIM_H, DIM_H);
    ln_relu_kernel<<<N_NODES / 8, 256, 0, stream>>>(h2, g2, b2, N_NODES);

    // 6. logits = h2@Wcls^T + bcls  (K1 = 0 path)
    gemm_wmma_kernel<<<dim3(gridM, 1), 256, 0, stream>>>(
        nullptr, h2, nullptr, Wcls, bcls, out, N_NODES, NUM_CLS, 0, DIM_H);
}